// SimGCL_Encoder_40931038331420
// MI455X (gfx1250) — compile-verified
//
#include <hip/hip_runtime.h>
#include <hip/hip_bf16.h>
#include <math.h>

#define USER_NUM 50000
#define ITEM_NUM 50000
#define NTOT     (USER_NUM + ITEM_NUM)   // 100000
#define EMB      128
#define NNZ_TOT  1600000
#define EPS_F    0.1f
#define EDGES_PER_WAVE 4

typedef __attribute__((ext_vector_type(2))) float v2f;
typedef __attribute__((ext_vector_type(8))) float v8f;

__device__ __forceinline__ float sgnf(float x) {
    return (x > 0.0f) ? 1.0f : ((x < 0.0f) ? -1.0f : 0.0f);
}

// X = concat(user_emb, item_emb); Y = 0; ACC (= d_out mean region) = 0
__global__ void init_kernel(const float4* __restrict__ U, const float4* __restrict__ I,
                            float4* __restrict__ X, float4* __restrict__ Y,
                            float4* __restrict__ ACC) {
    int i = blockIdx.x * blockDim.x + threadIdx.x;
    const int tot = NTOT * (EMB / 4);
    if (i >= tot) return;
    float4 v = (i < USER_NUM * (EMB / 4)) ? U[i] : I[i - USER_NUM * (EMB / 4)];
    X[i] = v;
    float4 z = make_float4(0.f, 0.f, 0.f, 0.f);
    Y[i] = z;
    ACC[i] = z;
}

// Row L2-norms of `noise` via WMMA Gram diagonal: one wave handles 16 rows.
// A = 16x4 f32 tile; with the f32 A layout, passing the same registers as B
// yields B = A^T, so D accumulates A*A^T; diag(D)[i] = sum_k noise[i,k]^2.
// Writes inv[row] = EPS / max(norm, 1e-12).
__global__ void rownorm_wmma_kernel(const float* __restrict__ noise,
                                    float* __restrict__ inv) {
    int wave = (blockIdx.x * blockDim.x + threadIdx.x) >> 5;
    int lane = threadIdx.x & 31;
    const int ngroups = NTOT / 16;   // 6250
    if (wave >= ngroups) return;     // whole-wave exit: EXEC all-ones at WMMA
    int rowBase = wave * 16;
    int m  = lane & 15;              // matrix row this lane supplies
    int kh = (lane >> 4) * 2;        // lanes 0-15: K=0,1 ; lanes 16-31: K=2,3
    const float* rp = noise + (size_t)(rowBase + m) * EMB + kh;

    v8f c = {};
#pragma unroll
    for (int kb = 0; kb < EMB; kb += 4) {
        v2f a = *(const v2f*)(rp + kb);     // 8B-aligned (kb + kh is even)
        c = __builtin_amdgcn_wmma_f32_16x16x4_f32(
                /*neg_a=*/false, a, /*neg_b=*/false, a,
                /*c_mod=*/(short)0, c, /*reuse_a=*/false, /*reuse_b=*/false);
    }
    // C/D layout: VGPR r holds M=r on lanes 0-15 (N=lane) and M=8+r on
    // lanes 16-31 (N=lane-16). Diagonal: (VGPR r, lane r) and (VGPR r, lane 24+r).
#pragma unroll
    for (int r = 0; r < 8; ++r) {
        float d  = c[r];
        float iv = EPS_F / fmaxf(sqrtf(d), 1e-12f);
        if (lane == r)      inv[rowBase + r]     = iv;
        if (lane == 24 + r) inv[rowBase + 8 + r] = iv;
    }
}

// COO SpMM: Y[row] += val * X[col]. One wave per edge-quad; 32 lanes * float4
// cover the full 128-wide embedding row. X/Y are L2-resident (51 MB each).
__global__ void spmm_kernel(const int* __restrict__ rows, const int* __restrict__ cols,
                            const float* __restrict__ vals,
                            const float4* __restrict__ X4, float* __restrict__ Y) {
    int wave = (blockIdx.x * blockDim.x + threadIdx.x) >> 5;
    int lane = threadIdx.x & 31;
    long e0 = (long)wave * EDGES_PER_WAVE;
    if (e0 >= NNZ_TOT) return;
    // stream-ahead prefetch of the COO arrays (global_prefetch_b8)
    __builtin_prefetch(rows + e0 + 4096, 0, 0);
    __builtin_prefetch(cols + e0 + 4096, 0, 0);
    __builtin_prefetch(vals + e0 + 4096, 0, 0);
#pragma unroll
    for (int t = 0; t < EDGES_PER_WAVE; ++t) {
        long e = e0 + t;
        int r  = rows[e];
        int cc = cols[e];
        float v = vals[e];
        float4 xv = X4[(size_t)cc * (EMB / 4) + lane];
        float* yp = Y + (size_t)r * EMB + lane * 4;
        atomicAdd(yp + 0, v * xv.x);
        atomicAdd(yp + 1, v * xv.y);
        atomicAdd(yp + 2, v * xv.z);
        atomicAdd(yp + 3, v * xv.w);
    }
}

// Fused post-SpMM for layers 0/1: ACC += Y; X = Y + sign(Y)*noise*inv[row];
// ACC += X; Y = 0 (ready for next layer).
__global__ void perturb_kernel(const float4* __restrict__ NZ4, const float* __restrict__ inv,
                               float4* __restrict__ X4, float4* __restrict__ Y4,
                               float4* __restrict__ ACC4) {
    int i = blockIdx.x * blockDim.x + threadIdx.x;
    const int tot = NTOT * (EMB / 4);
    if (i >= tot) return;
    int row = i >> 5;                 // EMB/4 == 32 float4 per row
    float s = inv[row];
    float4 y  = Y4[i];
    float4 nz = NZ4[i];
    float4 x;
    x.x = y.x + sgnf(y.x) * nz.x * s;
    x.y = y.y + sgnf(y.y) * nz.y * s;
    x.z = y.z + sgnf(y.z) * nz.z * s;
    x.w = y.w + sgnf(y.w) * nz.w * s;
    float4 a = ACC4[i];
    a.x += y.x + x.x;
    a.y += y.y + x.y;
    a.z += y.z + x.z;
    a.w += y.w + x.w;
    ACC4[i] = a;
    X4[i] = x;
    Y4[i] = make_float4(0.f, 0.f, 0.f, 0.f);
}

// Layer-2 epilogue: mean = (ACC + Y) / 5 ; neg = Y[items]
__global__ void final_kernel(const float4* __restrict__ Y4, float4* __restrict__ ACC4,
                             float4* __restrict__ NEG4) {
    int i = blockIdx.x * blockDim.x + threadIdx.x;
    const int tot = NTOT * (EMB / 4);
    if (i >= tot) return;
    float4 y = Y4[i];
    float4 a = ACC4[i];
    a.x = (a.x + y.x) * 0.2f;
    a.y = (a.y + y.y) * 0.2f;
    a.z = (a.z + y.z) * 0.2f;
    a.w = (a.w + y.w) * 0.2f;
    ACC4[i] = a;
    const int userTot = USER_NUM * (EMB / 4);
    if (i >= userTot) NEG4[i - userTot] = y;
}

extern "C" void kernel_launch(void* const* d_in, const int* in_sizes, int n_in,
                              void* d_out, int out_size, void* d_ws, size_t ws_size,
                              hipStream_t stream) {
    const float* user_emb = (const float*)d_in[0];
    const float* item_emb = (const float*)d_in[1];
    const int*   adj_rows = (const int*)d_in[2];
    const int*   adj_cols = (const int*)d_in[3];
    const float* adj_vals = (const float*)d_in[4];
    const float* noise0   = (const float*)d_in[5];
    const float* noise1   = (const float*)d_in[6];

    float* X    = (float*)d_ws;                       // N*128
    float* Y    = X + (size_t)NTOT * EMB;             // N*128
    float* inv0 = Y + (size_t)NTOT * EMB;             // N
    float* inv1 = inv0 + NTOT;                        // N
    float* ACC  = (float*)d_out;                      // N*128 (mean region)
    float* NEG  = ACC + (size_t)NTOT * EMB;           // ITEM*128

    const int tot4 = NTOT * (EMB / 4);                // 3.2M float4
    dim3 blk(256);
    dim3 grdE((tot4 + 255) / 256);

    init_kernel<<<grdE, blk, 0, stream>>>((const float4*)user_emb, (const float4*)item_emb,
                                          (float4*)X, (float4*)Y, (float4*)ACC);

    const int ngroups = NTOT / 16;                    // 6250 waves
    dim3 grdN((ngroups * 32 + 255) / 256);
    rownorm_wmma_kernel<<<grdN, blk, 0, stream>>>(noise0, inv0);
    rownorm_wmma_kernel<<<grdN, blk, 0, stream>>>(noise1, inv1);

    const long waves = NNZ_TOT / EDGES_PER_WAVE;      // 400000 waves
    dim3 grdS((unsigned)((waves * 32 + 255) / 256));

    // layer 0
    spmm_kernel<<<grdS, blk, 0, stream>>>(adj_rows, adj_cols, adj_vals, (const float4*)X, Y);
    perturb_kernel<<<grdE, blk, 0, stream>>>((const float4*)noise0, inv0,
                                             (float4*)X, (float4*)Y, (float4*)ACC);
    // layer 1
    spmm_kernel<<<grdS, blk, 0, stream>>>(adj_rows, adj_cols, adj_vals, (const float4*)X, Y);
    perturb_kernel<<<grdE, blk, 0, stream>>>((const float4*)noise1, inv1,
                                             (float4*)X, (float4*)Y, (float4*)ACC);
    // layer 2
    spmm_kernel<<<grdS, blk, 0, stream>>>(adj_rows, adj_cols, adj_vals, (const float4*)X, Y);
    final_kernel<<<grdE, blk, 0, stream>>>((const float4*)Y, (float4*)ACC, (float4*)NEG);
}